// inv_cg_22256520528520
// MI455X (gfx1250) — compile-verified
//
#include <hip/hip_runtime.h>

// ---------------------------------------------------------------------------
// Batched CG solve of (K^T K + rho I) x = y + rho*im on MI455X (gfx1250).
// 5x5 SAME correlation expressed as banded-Toeplitz WMMA f32 16x16x4 matmuls.
// All CG state lives in d_ws (fits in 192MB L2). Deterministic 2-stage
// per-image reductions (no float atomics) for graph-replay stability.
// B (Toeplitz) fragments are gathered branch-free from a zero-padded LDS
// kernel table once per wave; hot loop = 25 x (ds_load_b64 + v_wmma).
// ---------------------------------------------------------------------------

typedef __attribute__((ext_vector_type(2))) float v2f;
typedef __attribute__((ext_vector_type(8))) float v8f;

#define IMG_H   512
#define IMG_W   512
#define HW      (IMG_H * IMG_W)          // 262144 = 2^18
#define NIMG    16
#define NTOT    (NIMG * HW)              // 4194304
#define NTILES  (NIMG * 1024)            // 16 imgs * (32x32 tiles)

// workspace layout (float offsets)
#define WS_RHO      0
#define WS_RS       16
#define WS_ALPHA    48
#define WS_BETA     64
#define WS_PAP_PART 256                   // 16384 floats (one per wave/tile)
#define WS_RR_PART  16640                 // 16384 floats (one per block/tile)
#define WS_TMP      65536
#define WS_R        (WS_TMP + NTOT)
#define WS_P        (WS_R + NTOT)
#define WS_AP       (WS_P + NTOT)

__global__ void k_rho(const float* __restrict__ w, float* __restrict__ ws) {
    ws[WS_RHO] = 0.8f / (1.0f + __expf(-w[0]));
}

// One wave (32 lanes) computes one 16x16 output tile via 25 WMMA f32 16x16x4.
// MODE 0: out = corr(in, k)              -> ws[WS_TMP]            (conv1)
// MODE 1: r = p = y - corr(in, flip(k)); x = im; rr partials      (init)
// MODE 2: Ap = corr(in, flip(k)) + rho*p; p.Ap partials           (iter conv2)
template <int MODE>
__global__ __launch_bounds__(256) void k_conv(const float* __restrict__ in,
                                              const float* __restrict__ kern,
                                              float* __restrict__ ws,
                                              const float* __restrict__ y,
                                              const float* __restrict__ im,
                                              float* __restrict__ xout) {
    __shared__ float patch[8][400];   // 8 waves * 20x20 zero-padded patch
    __shared__ float skpad[200];      // 5 rows x 40, taps at [dy*40+15 .. +19]

    const int tid  = threadIdx.x;
    const int lane = tid & 31;
    const int wv   = tid >> 5;
    const int tile = blockIdx.x * 8 + wv;     // global tile id
    const int img  = tile >> 10;              // 1024 tiles per image
    const int tin  = tile & 1023;
    const int row0 = (tin >> 5) << 4;
    const int col0 = (tin & 31) << 4;

    // zero-padded kernel rows (flipped for MODE>=1): branch-free B gathers
    if (tid < 200) {
        const int dy = tid / 40;
        const int j  = tid - dy * 40;
        const int t  = j - 15;
        float v = 0.0f;
        if (t >= 0 && t < 5) {
            const int li = dy * 5 + t;
            v = (MODE == 0) ? kern[li] : kern[24 - li];
        }
        skpad[tid] = v;
    }

    const float* src = in + img * HW;
    for (int idx = lane; idx < 400; idx += 32) {
        const int rr = idx / 20;
        const int cc = idx - rr * 20;
        const int gr = row0 + rr - 2;
        const int gc = col0 + cc - 2;
        float v = 0.0f;
        if (gr >= 0 && gr < IMG_H && gc >= 0 && gc < IMG_W)
            v = src[gr * IMG_W + gc];
        patch[wv][idx] = v;
    }
    __syncthreads();

    const int h = lane >> 4;     // lane half
    const int n = lane & 15;     // A: row M ; B/C/D: column N

    // Pre-gather all 25 B fragments (identical for every tile) into registers.
    float bx[25], by[25];
    const int bbase = 2 * h - n + 15;
#pragma unroll
    for (int dy = 0; dy < 5; ++dy) {
#pragma unroll
        for (int q = 0; q < 5; ++q) {
            const int o = dy * 40 + 4 * q + bbase;
            bx[dy * 5 + q] = skpad[o];
            by[dy * 5 + q] = skpad[o + 1];
        }
    }

    v8f acc = {0.f, 0.f, 0.f, 0.f, 0.f, 0.f, 0.f, 0.f};

    // O(16x16) = sum_dy P_dy(16x20) * W_dy(20x16), K split into 5 chunks of 4.
#pragma unroll
    for (int dy = 0; dy < 5; ++dy) {
#pragma unroll
        for (int q = 0; q < 5; ++q) {
            // A fragment: lane holds M=n, K = 4q + 2h + {0,1}
            v2f a;
            const int pofs = (n + dy) * 20 + 4 * q + 2 * h;
            a.x = patch[wv][pofs];
            a.y = patch[wv][pofs + 1];
            // B fragment (Toeplitz of kernel row dy): K-row = 4q + 2h + v, col = n
            v2f b;
            b.x = bx[dy * 5 + q];
            b.y = by[dy * 5 + q];
            acc = __builtin_amdgcn_wmma_f32_16x16x4_f32(
                false, a, false, b, (short)0, acc, false, false);
        }
    }

    // C/D layout: VGPR v holds M = v + 8h, N = lane&15
    const int base = img * HW;
    if (MODE == 0) {
        float* tmp = ws + WS_TMP;
#pragma unroll
        for (int v = 0; v < 8; ++v) {
            const int i = base + (row0 + v + 8 * h) * IMG_W + (col0 + n);
            tmp[i] = acc[v];
        }
    } else if (MODE == 1) {
        float* r = ws + WS_R;
        float* p = ws + WS_P;
        float rr = 0.0f;
#pragma unroll
        for (int v = 0; v < 8; ++v) {
            const int i = base + (row0 + v + 8 * h) * IMG_W + (col0 + n);
            const float rv = y[i] - acc[v];
            r[i] = rv;
            p[i] = rv;
            xout[i] = im[i];
            rr += rv * rv;
        }
        for (int o = 16; o > 0; o >>= 1) rr += __shfl_xor(rr, o, 32);
        if (lane == 0) ws[WS_RR_PART + tile] = rr;
    } else {
        const float rho = ws[WS_RHO];
        const float* p = ws + WS_P;
        float* Ap = ws + WS_AP;
        float pap = 0.0f;
#pragma unroll
        for (int v = 0; v < 8; ++v) {
            const int i = base + (row0 + v + 8 * h) * IMG_W + (col0 + n);
            const float pv = p[i];
            const float av = acc[v] + rho * pv;
            Ap[i] = av;
            pap += pv * av;
        }
        for (int o = 16; o > 0; o >>= 1) pap += __shfl_xor(pap, o, 32);
        if (lane == 0) ws[WS_PAP_PART + tile] = pap;
    }
}

// x += alpha*p ; r -= alpha*Ap ; per-block r.r partials (1024 blocks per image)
__global__ __launch_bounds__(256) void k_update(float* __restrict__ x,
                                                float* __restrict__ ws) {
    __shared__ float sred[256];
    const int i = blockIdx.x * 256 + threadIdx.x;
    const int img = i >> 18;
    const float alpha = ws[WS_ALPHA + img];
    float* r = ws + WS_R;
    const float* p = ws + WS_P;
    const float* Ap = ws + WS_AP;
    x[i] += alpha * p[i];
    const float rv = r[i] - alpha * Ap[i];
    r[i] = rv;
    sred[threadIdx.x] = rv * rv;
    __syncthreads();
    for (int s = 128; s > 0; s >>= 1) {
        if (threadIdx.x < s) sred[threadIdx.x] += sred[threadIdx.x + s];
        __syncthreads();
    }
    if (threadIdx.x == 0) ws[WS_RR_PART + blockIdx.x] = sred[0];
}

// p = r + beta*p
__global__ __launch_bounds__(256) void k_pupd(float* __restrict__ ws) {
    const int i = blockIdx.x * 256 + threadIdx.x;
    const int img = i >> 18;
    const float beta = ws[WS_BETA + img];
    float* p = ws + WS_P;
    const float* r = ws + WS_R;
    p[i] = r[i] + beta * p[i];
}

// Deterministic second-stage reduction: 16 blocks, 1024 partials each.
// MODE 0: rs = sum       MODE 1: alpha = rs/(sum+eps)
// MODE 2: beta = sum/(rs+eps); rs = sum
template <int MODE>
__global__ __launch_bounds__(256) void k_reduce(float* __restrict__ ws, int partOfs) {
    __shared__ float sred[256];
    const int img = blockIdx.x;
    const float* part = ws + partOfs + img * 1024;
    float s = part[threadIdx.x] + part[threadIdx.x + 256] +
              part[threadIdx.x + 512] + part[threadIdx.x + 768];
    sred[threadIdx.x] = s;
    __syncthreads();
    for (int st = 128; st > 0; st >>= 1) {
        if (threadIdx.x < st) sred[threadIdx.x] += sred[threadIdx.x + st];
        __syncthreads();
    }
    if (threadIdx.x == 0) {
        const float tot = sred[0];
        if (MODE == 0) {
            ws[WS_RS + img] = tot;
        } else if (MODE == 1) {
            ws[WS_ALPHA + img] = ws[WS_RS + img] / (tot + 1e-12f);
        } else {
            ws[WS_BETA + img] = tot / (ws[WS_RS + img] + 1e-12f);
            ws[WS_RS + img] = tot;
        }
    }
}

extern "C" void kernel_launch(void* const* d_in, const int* in_sizes, int n_in,
                              void* d_out, int out_size, void* d_ws, size_t ws_size,
                              hipStream_t stream) {
    (void)in_sizes; (void)n_in; (void)out_size; (void)ws_size;
    const float* im   = (const float*)d_in[0];
    const float* y    = (const float*)d_in[1];
    const float* lw   = (const float*)d_in[2];
    const float* kern = (const float*)d_in[3];
    float* x  = (float*)d_out;
    float* ws = (float*)d_ws;

    const dim3 cgrid(NTILES / 8), cblk(256);

    k_rho<<<1, 1, 0, stream>>>(lw, ws);

    // r0 = p0 = y - conv2(conv1(im)); x0 = im; rs0 = sum r0^2
    k_conv<0><<<cgrid, cblk, 0, stream>>>(im, kern, ws, nullptr, nullptr, nullptr);
    k_conv<1><<<cgrid, cblk, 0, stream>>>(ws + WS_TMP, kern, ws, y, im, x);
    k_reduce<0><<<NIMG, 256, 0, stream>>>(ws, WS_RR_PART);

    for (int it = 0; it < 10; ++it) {
        k_conv<0><<<cgrid, cblk, 0, stream>>>(ws + WS_P, kern, ws, nullptr, nullptr, nullptr);
        k_conv<2><<<cgrid, cblk, 0, stream>>>(ws + WS_TMP, kern, ws, nullptr, nullptr, nullptr);
        k_reduce<1><<<NIMG, 256, 0, stream>>>(ws, WS_PAP_PART);
        k_update<<<NTOT / 256, 256, 0, stream>>>(x, ws);
        k_reduce<2><<<NIMG, 256, 0, stream>>>(ws, WS_RR_PART);
        k_pupd<<<NTOT / 256, 256, 0, stream>>>(ws);
    }
}